// GAT_24343874634357
// MI455X (gfx1250) — compile-verified
//
#include <hip/hip_runtime.h>

// ---------------------------------------------------------------------------
// Problem constants (from the reference)
// ---------------------------------------------------------------------------
#define NNODES 4096
#define NEDGES 262144
#define NFEAT  512
#define NHID   64
#define NHEAD  8
#define NCLASS 16
#define HCAT   (NHEAD * NHID)   // 512
#define GEMM_KMAX 512           // K of both GEMMs; B panel [K,16] staged in LDS (32 KB)

typedef __attribute__((ext_vector_type(2))) float    v2f;
typedef __attribute__((ext_vector_type(8))) float    v8f;
typedef __attribute__((ext_vector_type(4))) unsigned u32x4;
typedef __attribute__((ext_vector_type(4))) int      i32x4;
typedef __attribute__((ext_vector_type(8))) int      i32x8;

// ---------------------------------------------------------------------------
// Workspace layout (bytes), 256B aligned
// ---------------------------------------------------------------------------
__host__ __device__ constexpr size_t AL(size_t x) { return (x + 255) & ~(size_t)255; }

constexpr size_t OFF_WCAT   = 0;                                              // [512,512] f32
constexpr size_t OFF_H      = AL(OFF_WCAT   + (size_t)NFEAT * HCAT * 4);      // [N,512] f32
constexpr size_t OFF_X      = AL(OFF_H      + (size_t)NNODES * HCAT * 4);     // [N,512] f32
constexpr size_t OFF_PAL1   = AL(OFF_X      + (size_t)NNODES * HCAT * 4);     // [E,8]  f32
constexpr size_t OFF_BITMAP = AL(OFF_PAL1   + (size_t)NEDGES * NHEAD * 4);    // N*N bits
constexpr size_t OFF_VALID  = AL(OFF_BITMAP + (size_t)NNODES * NNODES / 8);   // [E] int
constexpr size_t OFF_EIDX   = AL(OFF_VALID  + (size_t)NEDGES * 4);            // [E] int
constexpr size_t OFF_PAL2   = AL(OFF_EIDX   + (size_t)NEDGES * 4);            // [E] f32
constexpr size_t OFF_DEG    = AL(OFF_PAL2   + (size_t)NEDGES * 4);            // [N] int
constexpr size_t OFF_ROWPTR = AL(OFF_DEG    + (size_t)NNODES * 4);            // [N+1] int
constexpr size_t OFF_FILL   = AL(OFF_ROWPTR + (size_t)(NNODES + 1) * 4);      // [N] int
constexpr size_t OFF_S1SRC  = AL(OFF_FILL   + (size_t)NNODES * 4);            // [N,8] f32
constexpr size_t OFF_S1DST  = AL(OFF_S1SRC  + (size_t)NNODES * NHEAD * 4);    // [N,8] f32
constexpr size_t OFF_HMEAN  = AL(OFF_S1DST  + (size_t)NNODES * NHEAD * 4);    // [512] f32
constexpr size_t OFF_H2     = AL(OFF_HMEAN  + (size_t)HCAT * 4);              // [N,16] f32
constexpr size_t OFF_S2SRC  = AL(OFF_H2     + (size_t)NNODES * NCLASS * 4);   // [N] f32
constexpr size_t OFF_S2DST  = AL(OFF_S2SRC  + (size_t)NNODES * 4);            // [N] f32
constexpr size_t OFF_H2MEAN = AL(OFF_S2DST  + (size_t)NNODES * 4);            // [16] f32
constexpr size_t WS_NEEDED  = AL(OFF_H2MEAN + (size_t)NCLASS * 4);

// ---------------------------------------------------------------------------
// Pack W_heads [8,512,64] -> Wcat [512, 512]  (Wcat[k][h*64+j] = W_heads[h][k][j])
// so that features @ Wcat lands directly in torch-concat layout.
// ---------------------------------------------------------------------------
__global__ void gat_pack_w(const float* __restrict__ Wh, float* __restrict__ Wcat) {
    int t = blockIdx.x * blockDim.x + threadIdx.x;      // t = k*512 + ncol
    if (t >= NFEAT * HCAT) return;
    int k = t >> 9, ncol = t & 511;
    int h = ncol >> 6, j = ncol & 63;
    Wcat[t] = Wh[(size_t)h * NFEAT * NHID + (size_t)k * NHID + j];
}

// ---------------------------------------------------------------------------
// FP32 WMMA GEMM: C[M,N] = A[M,K]*B[K,N], row-major. Block = 8 waves covering
// 128 rows x 16 cols; the 16-col B panel (K x 16) is staged once per block
// into LDS — via the Tensor Data Mover when available, else plain ds stores.
// One wave32 per 16x16 C tile, K stepped by 4 via V_WMMA_F32_16X16X4_F32.
// A frag (16x4): lane<16 -> M=lane, VGPR0/1=K 0/1 ; lane>=16 -> K 2/3.
// B frag (4x16): lane<16 -> N=lane, VGPR0/1=K 0/1 ; lane>=16 -> K 2/3.
// C/D (16x16): VGPR r -> M = r + 8*(lane>=16), N = lane&15.
// grid = (N/16, M/128)
// ---------------------------------------------------------------------------
__global__ __launch_bounds__(256) void gat_gemm_wmma_f32(
    const float* __restrict__ A, const float* __restrict__ B, float* __restrict__ C,
    int M, int N, int K)
{
    __shared__ float Bs[GEMM_KMAX * 16];                // 32 KB B panel

    const int nt     = blockIdx.x;                      // n-tile (16 cols)
    const int ntbase = nt * 16;
    const int mtile  = blockIdx.y * 8 + (threadIdx.x >> 5);
    const int lane   = threadIdx.x & 31;
    const int r      = lane & 15;
    const int kk     = (lane >> 4) << 1;                // 0 or 2

    // ---- stage B[K,16] panel into LDS --------------------------------------
#if __has_builtin(__builtin_amdgcn_tensor_load_to_lds) && \
    __has_builtin(__builtin_amdgcn_s_wait_tensorcnt)
    if ((threadIdx.x >> 5) == 0) {
        // Tensor DMA descriptor (D#), ISA 8.3-8.6: 2D tile, 4B elements,
        // tile 16 x K at global_addr = &B[0*N + ntbase], row stride N.
        unsigned long long ga = (unsigned long long)(uintptr_t)(B + ntbase);
        unsigned lds = (unsigned)(uintptr_t)(&Bs[0]);   // low 32 bits = LDS byte addr
        u32x4 g0;
        g0[0] = 1u;                                         // count=1, user mode
        g0[1] = lds;                                        // lds_addr
        g0[2] = (unsigned)(ga & 0xFFFFFFFFu);               // global_addr[31:0]
        g0[3] = (unsigned)((ga >> 32) & 0x01FFFFFFu)        // global_addr[56:32]
              | (2u << 30);                                 // type = 2 ("image")
        i32x8 g1;
        g1[0] = (int)(2u << 16);                            // data_size = 4B, mask=0
        g1[1] = (int)(((unsigned)N & 0xFFFFu) << 16);       // tensor_dim0[15:0]
        g1[2] = (int)((((unsigned)N >> 16) & 0xFFFFu)       // tensor_dim0[31:16]
              | (((unsigned)K & 0xFFFFu) << 16));           // tensor_dim1[15:0]
        g1[3] = (int)((((unsigned)K >> 16) & 0xFFFFu)       // tensor_dim1[31:16]
              | (16u << 16));                               // tile_dim0 = 16
        g1[4] = (int)((unsigned)K & 0xFFFFu);               // tile_dim1 = K, tile_dim2=0
        g1[5] = (int)N;                                     // tensor_dim0_stride[31:0]
        g1[6] = 0;                                          // stride hi, dim1_stride lo
        g1[7] = 0;
        i32x4 z4 = {0, 0, 0, 0};
#if defined(__clang_major__) && (__clang_major__ >= 23)
        i32x8 z8 = {0, 0, 0, 0, 0, 0, 0, 0};
        __builtin_amdgcn_tensor_load_to_lds(g0, g1, z4, z4, z8, 0);
#else
        __builtin_amdgcn_tensor_load_to_lds(g0, g1, z4, z4, 0);
#endif
        __builtin_amdgcn_s_wait_tensorcnt(0);
    }
#else
    for (int idx = threadIdx.x; idx < K * 16; idx += 256) {
        int k = idx >> 4, c = idx & 15;
        Bs[idx] = B[(size_t)k * N + ntbase + c];
    }
#endif
    __syncthreads();

    // ---- WMMA main loop -----------------------------------------------------
    const float* ap = A + (size_t)(mtile * 16 + r) * K + kk;
    const float* bp = Bs + kk * 16 + r;

    v8f acc = {};
    for (int k0 = 0; k0 < K; k0 += 16) {
        __builtin_prefetch(ap + 64, 0, 1);              // gfx1250 global_prefetch path
#pragma unroll
        for (int u = 0; u < 4; ++u) {
            v2f a = *(const v2f*)(ap + u * 4);          // A[m, k0+u*4+kk .. +1]
            v2f b;
            b[0] = bp[(u * 4) * 16];                    // Bs[k0+u*4+kk  ][r]
            b[1] = bp[(u * 4 + 1) * 16];                // Bs[k0+u*4+kk+1][r]
            acc = __builtin_amdgcn_wmma_f32_16x16x4_f32(
                false, a, false, b, (short)0, acc, false, false);
        }
        ap += 16;
        bp += 16 * 16;
    }

    const int col     = ntbase + r;
    const int rowHalf = (lane >> 4) << 3;               // 0 or 8
#pragma unroll
    for (int i = 0; i < 8; ++i)
        C[(size_t)(mtile * 16 + rowHalf + i) * N + col] = acc[i];
}

// ---------------------------------------------------------------------------
// Layer-1 per-node attention half-scores: s_src[i,h]=h_i.a[h,:64], s_dst=h_i.a[h,64:]
// ---------------------------------------------------------------------------
__global__ void gat_node_scores1(const float* __restrict__ H, const float* __restrict__ a,
                                 float* __restrict__ ssrc, float* __restrict__ sdst) {
    int t = blockIdx.x * blockDim.x + threadIdx.x;  // t = i*8 + h
    if (t >= NNODES * NHEAD) return;
    int i = t >> 3, h = t & 7;
    const float* hp = H + (size_t)i * HCAT + h * NHID;
    const float* ap = a + h * 2 * NHID;
    float s = 0.f, d = 0.f;
#pragma unroll
    for (int j = 0; j < NHID; ++j) {
        s = fmaf(hp[j], ap[j], s);
        d = fmaf(hp[j], ap[NHID + j], d);
    }
    ssrc[t] = s; sdst[t] = d;
}

// Dedupe duplicate (src,tgt) cells (reference .set keeps one; dup scores identical)
__global__ void gat_dedupe(const int* __restrict__ el, unsigned* __restrict__ bm,
                           int* __restrict__ valid, int* __restrict__ deg) {
    int e = blockIdx.x * blockDim.x + threadIdx.x;
    if (e >= NEDGES) return;
    int s = el[e], d = el[NEDGES + e];
    unsigned key = (unsigned)s * (unsigned)NNODES + (unsigned)d;
    unsigned bit = 1u << (key & 31);
    unsigned old = atomicOr(&bm[key >> 5], bit);
    if (!(old & bit)) { valid[e] = 1; atomicAdd(&deg[s], 1); }
    else              { valid[e] = 0; }
}

__global__ void gat_scan(const int* __restrict__ deg, int* __restrict__ rowptr) {
    if (blockIdx.x == 0 && threadIdx.x == 0) {
        int run = 0;
        for (int i = 0; i < NNODES; ++i) { rowptr[i] = run; run += deg[i]; }
        rowptr[NNODES] = run;
    }
}

__global__ void gat_scatter(const int* __restrict__ el, const int* __restrict__ valid,
                            const int* __restrict__ rowptr, int* __restrict__ fill,
                            int* __restrict__ eidx) {
    int e = blockIdx.x * blockDim.x + threadIdx.x;
    if (e >= NEDGES || !valid[e]) return;
    int s = el[e];
    int pos = atomicAdd(&fill[s], 1);
    eidx[rowptr[s] + pos] = e;
}

// Stable per-row ordering (insertion sort by edge id) -> deterministic reductions
__global__ void gat_sortrows(const int* __restrict__ rowptr, int* __restrict__ eidx) {
    int i = blockIdx.x * blockDim.x + threadIdx.x;
    if (i >= NNODES) return;
    int s = rowptr[i], e = rowptr[i + 1];
    for (int a = s + 1; a < e; ++a) {
        int v = eidx[a]; int b = a - 1;
        while (b >= s && eidx[b] > v) { eidx[b + 1] = eidx[b]; --b; }
        eidx[b + 1] = v;
    }
}

__global__ void gat_edge_scores1(const int* __restrict__ el, const float* __restrict__ ssrc,
                                 const float* __restrict__ sdst, float* __restrict__ pal) {
    int t = blockIdx.x * blockDim.x + threadIdx.x;  // t = e*8 + h
    if (t >= NEDGES * NHEAD) return;
    int e = t >> 3, h = t & 7;
    int s = el[e], d = el[NEDGES + e];
    float v = ssrc[s * NHEAD + h] + sdst[d * NHEAD + h];
    pal[t] = v > 0.f ? v : 0.2f * v;                // LeakyReLU(0.2)
}

__global__ void gat_edge_scores2(const int* __restrict__ el, const float* __restrict__ ssrc,
                                 const float* __restrict__ sdst, float* __restrict__ pal) {
    int e = blockIdx.x * blockDim.x + threadIdx.x;
    if (e >= NEDGES) return;
    int s = el[e], d = el[NEDGES + e];
    float v = ssrc[s] + sdst[d];
    pal[e] = v > 0.f ? v : 0.2f * v;
}

// One wave32 per (row, head): max -> sum(exp) -> normalize, in-place on pal.
__global__ void gat_row_softmax(const int* __restrict__ rowptr, const int* __restrict__ eidx,
                                float* __restrict__ pal, int nheads, int nrows) {
    int w = blockIdx.x * (blockDim.x >> 5) + (threadIdx.x >> 5);
    if (w >= nrows * nheads) return;
    int row = w / nheads, h = w - row * nheads;
    int lane = threadIdx.x & 31;
    int s = rowptr[row], e = rowptr[row + 1];

    float m = -3.0e38f;
    for (int p = s + lane; p < e; p += 32)
        m = fmaxf(m, pal[eidx[p] * nheads + h]);
#pragma unroll
    for (int off = 16; off; off >>= 1) m = fmaxf(m, __shfl_xor(m, off, 32));

    float sum = 0.f;
    for (int p = s + lane; p < e; p += 32)
        sum += expf(pal[eidx[p] * nheads + h] - m);
#pragma unroll
    for (int off = 16; off; off >>= 1) sum += __shfl_xor(sum, off, 32);

    float inv = 1.0f / sum;                          // empty row: never read below
    for (int p = s + lane; p < e; p += 32) {
        int idx = eidx[p] * nheads + h;
        pal[idx] = expf(pal[idx] - m) * inv;
    }
}

__global__ void gat_colmean(const float* __restrict__ H, float* __restrict__ mean,
                            int rows, int cols) {
    int j = blockIdx.x * blockDim.x + threadIdx.x;
    if (j >= cols) return;
    float s = 0.f;
    for (int i = 0; i < rows; ++i) s += H[(size_t)i * cols + j];
    mean[j] = s / (float)rows;
}

// Layer-1 aggregation + bias + ELU: one block (512 thr = h*64+j) per row.
__global__ __launch_bounds__(512) void gat_aggregate1(
    const int* __restrict__ rowptr, const int* __restrict__ eidx,
    const int* __restrict__ el, const float* __restrict__ pal,
    const float* __restrict__ H, const float* __restrict__ Hmean,
    const float* __restrict__ b_heads, float* __restrict__ X)
{
    int row = blockIdx.x;
    int tid = threadIdx.x;                           // tid = h*64+j
    int h = tid >> 6;
    int s = rowptr[row], e = rowptr[row + 1];
    float acc = 0.f;
    for (int p = s; p < e; ++p) {
        int ed = eidx[p];
        int t  = el[NEDGES + ed];
        acc = fmaf(pal[ed * NHEAD + h], H[(size_t)t * HCAT + tid], acc);
    }
    if (s == e) acc = Hmean[tid];                    // empty row -> uniform softmax
    float z = acc + b_heads[tid];                    // b_heads flat [8*64] matches tid
    X[(size_t)row * HCAT + tid] = z > 0.f ? z : (expf(z) - 1.0f);   // ELU
}

__global__ void gat_node_scores2(const float* __restrict__ H2, const float* __restrict__ a,
                                 float* __restrict__ ssrc, float* __restrict__ sdst) {
    int i = blockIdx.x * blockDim.x + threadIdx.x;
    if (i >= NNODES) return;
    float s = 0.f, d = 0.f;
#pragma unroll
    for (int c = 0; c < NCLASS; ++c) {
        float v = H2[i * NCLASS + c];
        s = fmaf(v, a[c], s);
        d = fmaf(v, a[NCLASS + c], d);
    }
    ssrc[i] = s; sdst[i] = d;
}

// Layer-2 aggregation + bias + log_softmax: one wave per row, lanes&15 = class.
__global__ __launch_bounds__(128) void gat_aggregate2(
    const int* __restrict__ rowptr, const int* __restrict__ eidx,
    const int* __restrict__ el, const float* __restrict__ pal2,
    const float* __restrict__ H2, const float* __restrict__ h2mean,
    const float* __restrict__ b_out, float* __restrict__ out)
{
    int row = blockIdx.x * (blockDim.x >> 5) + (threadIdx.x >> 5);
    if (row >= NNODES) return;
    int lane = threadIdx.x & 31;
    int c = lane & 15;
    int s = rowptr[row], e = rowptr[row + 1];
    float acc = 0.f;
    for (int p = s; p < e; ++p) {
        int ed = eidx[p];
        int t  = el[NEDGES + ed];
        acc = fmaf(pal2[ed], H2[t * NCLASS + c], acc);
    }
    if (s == e) acc = h2mean[c];
    float v = acc + b_out[c];
    float m = v;
#pragma unroll
    for (int off = 8; off; off >>= 1) m = fmaxf(m, __shfl_xor(m, off, 16));
    float se = expf(v - m);
#pragma unroll
    for (int off = 8; off; off >>= 1) se += __shfl_xor(se, off, 16);
    if (lane < 16) out[row * NCLASS + c] = v - m - logf(se);
}

// ---------------------------------------------------------------------------
extern "C" void kernel_launch(void* const* d_in, const int* in_sizes, int n_in,
                              void* d_out, int out_size, void* d_ws, size_t ws_size,
                              hipStream_t stream) {
    (void)in_sizes; (void)n_in; (void)out_size;
    if (ws_size < WS_NEEDED) return;

    const float* features = (const float*)d_in[0];
    const int*   el       = (const int*)  d_in[1];   // [2,E]: src = el[e], tgt = el[E+e]
    const float* W_heads  = (const float*)d_in[2];
    const float* a_heads  = (const float*)d_in[3];
    const float* b_heads  = (const float*)d_in[4];
    const float* W_out    = (const float*)d_in[5];   // [512,16] row-major
    const float* a_out    = (const float*)d_in[6];
    const float* b_out    = (const float*)d_in[7];
    float* out = (float*)d_out;

    char* ws = (char*)d_ws;
    float*    Wcat   = (float*)   (ws + OFF_WCAT);
    float*    H      = (float*)   (ws + OFF_H);
    float*    X      = (float*)   (ws + OFF_X);
    float*    pal1   = (float*)   (ws + OFF_PAL1);
    unsigned* bitmap = (unsigned*)(ws + OFF_BITMAP);
    int*      valid  = (int*)     (ws + OFF_VALID);
    int*      eidx   = (int*)     (ws + OFF_EIDX);
    float*    pal2   = (float*)   (ws + OFF_PAL2);
    int*      deg    = (int*)     (ws + OFF_DEG);
    int*      rowptr = (int*)     (ws + OFF_ROWPTR);
    int*      fill   = (int*)     (ws + OFF_FILL);
    float*    s1src  = (float*)   (ws + OFF_S1SRC);
    float*    s1dst  = (float*)   (ws + OFF_S1DST);
    float*    Hmean  = (float*)   (ws + OFF_HMEAN);
    float*    H2     = (float*)   (ws + OFF_H2);
    float*    s2src  = (float*)   (ws + OFF_S2SRC);
    float*    s2dst  = (float*)   (ws + OFF_S2DST);
    float*    h2mean = (float*)   (ws + OFF_H2MEAN);

    // zero-init (graph-capture-legal async memsets)
    hipMemsetAsync(bitmap, 0, (size_t)NNODES * NNODES / 8, stream);
    hipMemsetAsync(deg,    0, (size_t)NNODES * 4, stream);
    hipMemsetAsync(fill,   0, (size_t)NNODES * 4, stream);

    // ---- Layer 1 ----
    gat_pack_w<<<(NFEAT * HCAT + 255) / 256, 256, 0, stream>>>(W_heads, Wcat);

    {   // H = features @ Wcat  (4096x512x512, fp32 WMMA, TDM-staged B panel)
        dim3 grid(HCAT / 16, NNODES / 128);
        gat_gemm_wmma_f32<<<grid, 256, 0, stream>>>(features, Wcat, H,
                                                    NNODES, HCAT, NFEAT);
    }

    gat_node_scores1<<<(NNODES * NHEAD + 255) / 256, 256, 0, stream>>>(H, a_heads, s1src, s1dst);
    gat_dedupe<<<(NEDGES + 255) / 256, 256, 0, stream>>>(el, bitmap, valid, deg);
    gat_scan<<<1, 32, 0, stream>>>(deg, rowptr);
    gat_scatter<<<(NEDGES + 255) / 256, 256, 0, stream>>>(el, valid, rowptr, fill, eidx);
    gat_sortrows<<<(NNODES + 255) / 256, 256, 0, stream>>>(rowptr, eidx);
    gat_edge_scores1<<<(NEDGES * NHEAD + 255) / 256, 256, 0, stream>>>(el, s1src, s1dst, pal1);
    gat_row_softmax<<<(NNODES * NHEAD + 7) / 8, 256, 0, stream>>>(rowptr, eidx, pal1, NHEAD, NNODES);
    gat_colmean<<<(HCAT + 255) / 256, 256, 0, stream>>>(H, Hmean, NNODES, HCAT);
    gat_aggregate1<<<NNODES, 512, 0, stream>>>(rowptr, eidx, el, pal1, H, Hmean, b_heads, X);

    // ---- Layer 2 ----
    {   // H2 = X @ W_out  (4096x16x512, fp32 WMMA)
        dim3 grid(NCLASS / 16, NNODES / 128);
        gat_gemm_wmma_f32<<<grid, 256, 0, stream>>>(X, W_out, H2,
                                                    NNODES, NCLASS, HCAT);
    }
    gat_node_scores2<<<(NNODES + 255) / 256, 256, 0, stream>>>(H2, a_out, s2src, s2dst);
    gat_edge_scores2<<<(NEDGES + 255) / 256, 256, 0, stream>>>(el, s2src, s2dst, pal2);
    gat_row_softmax<<<(NNODES + 7) / 8, 256, 0, stream>>>(rowptr, eidx, pal2, 1, NNODES);
    gat_colmean<<<1, 32, 0, stream>>>(H2, h2mean, NNODES, NCLASS);
    gat_aggregate2<<<(NNODES + 3) / 4, 128, 0, stream>>>(rowptr, eidx, el, pal2, H2, h2mean,
                                                         b_out, out);
}